// AE_CSNMF_VQ_only_40819369181838
// MI455X (gfx1250) — compile-verified
//
#include <hip/hip_runtime.h>

typedef __attribute__((ext_vector_type(2))) float v2f;
typedef __attribute__((ext_vector_type(8))) float v8f;

// Problem constants
constexpr int Bc   = 16;
constexpr int Pc   = 12;
constexpr int Tc   = 4096;
constexpr int WINc = 41;
constexpr int PADc = 20;
constexpr int Kc   = 1024;
constexpr int Dc   = Pc * WINc;      // 492
constexpr int KSTEPS = Dc / 4;       // 123
constexpr int NT     = 16;           // 16x16 N-tiles per wave -> 256 columns
constexpr int ROWS_PER_BLK = 32;
constexpr int XW    = ROWS_PER_BLK + WINc - 1;   // 72: staged window width
constexpr int NBLK  = (Bc * Tc) / ROWS_PER_BLK;  // 2048
constexpr float SCALE = 0.25f / (float)(Bc * Tc * Dc);

// ---------------------------------------------------------------------------
// Kernel 1: cnorm[n] = sum_d codebook[n][d]^2
// ---------------------------------------------------------------------------
__global__ void __launch_bounds__(128)
cnorm_kernel(const float* __restrict__ cb, float* __restrict__ cnorm) {
    const int n = blockIdx.x;
    const float* row = cb + n * Dc;
    float s = 0.0f;
    for (int d = threadIdx.x; d < Dc; d += 128) {
        float v = row[d];
        s += v * v;
    }
    __shared__ float red[128];
    red[threadIdx.x] = s;
    __syncthreads();
    for (int off = 64; off > 0; off >>= 1) {
        if (threadIdx.x < off) red[threadIdx.x] += red[threadIdx.x + off];
        __syncthreads();
    }
    if (threadIdx.x == 0) cnorm[n] = red[0];
}

// ---------------------------------------------------------------------------
// Kernel 2: fused windowed-GEMM (fp32 WMMA) + row-min + block partial sum.
// Block = 256 threads = 8 waves. Block owns 32 consecutive rows (b,t).
// wave = rg*4 + cw : rg = row-group (16 rows each), cw = column quarter (256 N).
// A-fragments come from an LDS-staged x window; B-fragments are float2 global
// loads from the (L2-resident) codebook at compile-time constant offsets.
// ---------------------------------------------------------------------------
__global__ void __launch_bounds__(256)
vq_dist_kernel(const float* __restrict__ x, const float* __restrict__ cb,
               const float* __restrict__ cnorm, float* __restrict__ partial) {
    const int tid    = threadIdx.x;
    const int wave   = tid >> 5;
    const int lane   = tid & 31;
    const int rg     = wave >> 2;        // 0..1
    const int cw     = wave & 3;         // 0..3
    const int half   = lane >> 4;        // 0..1 (K sub-pair for A/B fragments)
    const int lane15 = lane & 15;

    const int rowbase = blockIdx.x * ROWS_PER_BLK;
    const int b       = rowbase >> 12;           // rowbase / T
    const int tbase   = rowbase & (Tc - 1);
    const float* xb   = x + (size_t)b * (Pc * Tc);

    // ---- stage zero-padded x window for this block's 32 rows into LDS ----
    __shared__ float xwin[Pc * XW];              // 12 x 72 floats = 3456 B
    for (int i = tid; i < Pc * XW; i += 256) {
        const int p  = i / XW;
        const int o  = i - p * XW;
        const int tt = tbase + o - PADc;
        xwin[i] = (tt >= 0 && tt < Tc) ? xb[p * Tc + tt] : 0.0f;
    }
    __syncthreads();

    const int tloc = rg * 16 + lane15;           // A-row (local) for this lane
    const int ncol = cw * (NT * 16) + lane15;    // per-lane base N column

    v8f acc[NT];
    const v8f vzero = {};
    #pragma unroll
    for (int i = 0; i < NT; ++i) acc[i] = vzero;

    float sq = 0.0f;   // ||flat_row||^2 partial (only cw==0 waves keep it)

    // Incremental (p, w) tracking for flat index k = ks*4 + half*2 (+0/+1):
    // w advances by 4 each step; on wrap past WIN, hop to next channel row.
    int w0   = half * 2;
    int w1   = half * 2 + 1;
    int idx0 = tloc + w0;                        // p0*XW + tloc + w0, p0 = 0
    int idx1 = tloc + w1;

    const float* bptr = cb + (size_t)ncol * Dc + half * 2;

    for (int ks = 0; ks < KSTEPS; ++ks) {
        const float a0 = xwin[idx0];
        const float a1 = xwin[idx1];
        v2f afrag;
        afrag.x = a0;
        afrag.y = a1;
        if (cw == 0) sq += a0 * a0 + a1 * a1;

        #pragma unroll
        for (int tle = 0; tle < NT; ++tle) {
            const v2f bfrag = *(const v2f*)(bptr + tle * (16 * Dc));
            acc[tle] = __builtin_amdgcn_wmma_f32_16x16x4_f32(
                false, afrag, false, bfrag, (short)0, acc[tle], false, false);
        }

        bptr += 4;
        w0 += 4; idx0 += 4; if (w0 >= WINc) { w0 -= WINc; idx0 += XW - WINc; }
        w1 += 4; idx1 += 4; if (w1 >= WINc) { w1 -= WINc; idx1 += XW - WINc; }
    }

    // --- per-lane cnorm values for this wave's columns ---
    float cn[NT];
    #pragma unroll
    for (int tle = 0; tle < NT; ++tle) cn[tle] = cnorm[ncol + tle * 16];

    __shared__ float valsLDS[ROWS_PER_BLK][4];
    __shared__ float xnormLDS[ROWS_PER_BLK];
    __shared__ float contribLDS[ROWS_PER_BLK];

    // C/D layout: VGPR j holds M = j (lanes 0-15, N=lane) and M = j+8 (lanes 16-31).
    #pragma unroll
    for (int j = 0; j < 8; ++j) {
        float mv = 3.0e38f;
        #pragma unroll
        for (int tle = 0; tle < NT; ++tle) {
            const float score = cn[tle] - 2.0f * acc[tle][j];
            mv = fminf(mv, score);
        }
        // min across the 16 lanes of each half (masks < 16 stay within halves)
        #pragma unroll
        for (int off = 1; off < 16; off <<= 1)
            mv = fminf(mv, __shfl_xor(mv, off, 32));
        if (lane15 == 0)
            valsLDS[rg * 16 + j + half * 8][cw] = mv;
    }

    if (cw == 0) {
        // lane L covers K%4 in {0,1}; lane L^16 covers {2,3} -> combine halves
        sq += __shfl_xor(sq, 16, 32);
        if (half == 0) xnormLDS[rg * 16 + lane15] = sq;
    }

    __syncthreads();

    if (tid < ROWS_PER_BLK) {
        const float mv = fminf(fminf(valsLDS[tid][0], valsLDS[tid][1]),
                               fminf(valsLDS[tid][2], valsLDS[tid][3]));
        contribLDS[tid] = mv + xnormLDS[tid];   // = min_n ||flat - cb_n||^2
    }
    __syncthreads();

    if (tid == 0) {
        float s = 0.0f;
        #pragma unroll
        for (int i = 0; i < ROWS_PER_BLK; ++i) s += contribLDS[i];
        partial[blockIdx.x] = s;
    }
}

// ---------------------------------------------------------------------------
// Kernel 3: deterministic fixed-order reduction of block partials
// ---------------------------------------------------------------------------
__global__ void __launch_bounds__(256)
finalize_kernel(const float* __restrict__ partial, float* __restrict__ out) {
    __shared__ float red[256];
    float s = 0.0f;
    for (int i = threadIdx.x; i < NBLK; i += 256) s += partial[i];
    red[threadIdx.x] = s;
    __syncthreads();
    for (int off = 128; off > 0; off >>= 1) {
        if (threadIdx.x < off) red[threadIdx.x] += red[threadIdx.x + off];
        __syncthreads();
    }
    if (threadIdx.x == 0) out[0] = red[0] * SCALE;
}

// ---------------------------------------------------------------------------
extern "C" void kernel_launch(void* const* d_in, const int* in_sizes, int n_in,
                              void* d_out, int out_size, void* d_ws, size_t ws_size,
                              hipStream_t stream) {
    const float* x  = (const float*)d_in[0];   // (B, P, T) f32
    const float* cb = (const float*)d_in[1];   // (K, D)    f32
    float* out      = (float*)d_out;           // scalar f32

    float* cnorm   = (float*)d_ws;             // K floats
    float* partial = cnorm + Kc;               // NBLK floats

    cnorm_kernel<<<Kc, 128, 0, stream>>>(cb, cnorm);
    vq_dist_kernel<<<NBLK, 256, 0, stream>>>(x, cb, cnorm, partial);
    finalize_kernel<<<1, 256, 0, stream>>>(partial, out);
}